// SGSPAssign2_18537078849989
// MI455X (gfx1250) — compile-verified
//
#include <hip/hip_runtime.h>

typedef __attribute__((ext_vector_type(2))) float v2f;
typedef __attribute__((ext_vector_type(8))) float v8f;

#define DDIM 512
#define HDIM 256
#define KDIM 64

// ---------------- degree / symmetric norm ----------------
__global__ __launch_bounds__(256) void k_init_deg(float* deg, int n) {
  int i = blockIdx.x * 256 + threadIdx.x;
  if (i < n) deg[i] = 1.0f;  // self-loop contributes 1 to every node's degree
}

__global__ __launch_bounds__(256) void k_deg_accum(const long long* __restrict__ dst,
                                                   float* deg, long long e) {
  long long i = (long long)blockIdx.x * 256 + threadIdx.x;
  if (i < e) unsafeAtomicAdd(&deg[(int)dst[i]], 1.0f);
}

__global__ __launch_bounds__(256) void k_dinv(float* deg, int n) {
  int i = blockIdx.x * 256 + threadIdx.x;
  if (i < n) {
    float d = deg[i];
    deg[i] = d > 0.0f ? rsqrtf(d) : 0.0f;  // in place: deg -> dinv
  }
}

// ---------------- fp32 WMMA GEMM: C[M,NC] = op(A[M,KD]) @ B[KD,NC] ----------------
// block = 256 threads = 8 wave32; block tile = 128 x 64; wave tile = 16 x 64.
// Double-buffered LDS staging (global float4 loads overlap with WMMA compute).
// B stored K-pair-packed in LDS so each WMMA B operand is one aligned ds_load_b64.
template <int KD, int NC, bool RELU>
__global__ __launch_bounds__(256) void k_gemm_wmma(const float* __restrict__ A,
                                                   const float* __restrict__ B,
                                                   float* __restrict__ C, int M) {
  __shared__ float As[2][128 * 64];  // 2 x 32 KB, row-major [r][k]
  __shared__ float Bs[2][64 * 64];   // 2 x 16 KB, packed [k/2][n][k&1]
  constexpr int NCHUNK = KD / 64;

  const int tid  = threadIdx.x;
  const int wave = tid >> 5;
  const int lane = tid & 31;
  const int hi   = lane >> 4;   // 0: K pair {0,1}; 1: K pair {2,3}
  const int nl   = lane & 15;   // M row (A) / N col (B,D) within tile
  const int bm   = blockIdx.x * 128;
  const int n0   = blockIdx.y * 64;
  const int m0   = bm + wave * 16;
  const bool active = (m0 < M);  // M is a multiple of 16, so active strips are full

  v8f acc[4] = {};
  float4 ra[8];  // staged A: 8 float4 per thread (8192 floats/chunk)
  float4 rb[4];  // staged B: 4 float4 per thread (4096 floats/chunk)

  auto load_chunk = [&](int kc) {
#pragma unroll
    for (int j = 0; j < 8; ++j) {
      int idx4 = tid + j * 256;
      int r = idx4 >> 4;
      int kk = (idx4 & 15) * 4;
      float4 v = make_float4(0.f, 0.f, 0.f, 0.f);
      if (bm + r < M) v = *(const float4*)&A[(size_t)(bm + r) * KD + (kc + kk)];
      if (RELU) {
        v.x = fmaxf(v.x, 0.f); v.y = fmaxf(v.y, 0.f);
        v.z = fmaxf(v.z, 0.f); v.w = fmaxf(v.w, 0.f);
      }
      ra[j] = v;
    }
#pragma unroll
    for (int j = 0; j < 4; ++j) {
      int idx4 = tid + j * 256;
      int kk = idx4 >> 4;
      int nn = (idx4 & 15) * 4;
      rb[j] = *(const float4*)&B[(size_t)(kc + kk) * NC + (n0 + nn)];
    }
  };

  auto store_chunk = [&](int buf) {
#pragma unroll
    for (int j = 0; j < 8; ++j) {
      int idx4 = tid + j * 256;
      *(float4*)&As[buf][idx4 * 4] = ra[j];
    }
#pragma unroll
    for (int j = 0; j < 4; ++j) {
      int idx4 = tid + j * 256;
      int kk = idx4 >> 4;
      int nn = (idx4 & 15) * 4;
      float* bp = &Bs[buf][(kk >> 1) * 128 + nn * 2 + (kk & 1)];
      bp[0] = rb[j].x; bp[2] = rb[j].y; bp[4] = rb[j].z; bp[6] = rb[j].w;
    }
  };

  auto compute = [&](int buf) {
    const float* as = &As[buf][(wave * 16 + nl) * 64];
    const float* bs = &Bs[buf][hi * 128 + nl * 2];
#pragma unroll
    for (int kk = 0; kk < 64; kk += 4) {
      const int ka = kk + 2 * hi;
      // A 16x4 layout: lanes 0-15 -> K={0,1}, lanes 16-31 -> K={2,3}, M = nl
      v2f a = *(const v2f*)(as + ka);
#pragma unroll
      for (int t = 0; t < 4; ++t) {
        // packed B: {B[ka][n], B[ka+1][n]} contiguous -> single ds_load_b64
        v2f b = *(const v2f*)(bs + (kk >> 1) * 128 + t * 32);
        acc[t] = __builtin_amdgcn_wmma_f32_16x16x4_f32(
            false, a, false, b, (short)0, acc[t], false, false);
      }
    }
  };

  load_chunk(0);
  store_chunk(0);
  __syncthreads();
  for (int c = 0; c < NCHUNK; ++c) {
    if (c + 1 < NCHUNK) load_chunk((c + 1) * 64);  // overlap with compute below
    if (active) compute(c & 1);
    if (c + 1 < NCHUNK) {
      __syncthreads();              // everyone done reading buffer (c+1)&1
      store_chunk((c + 1) & 1);
      __syncthreads();              // stores visible before next compute
    }
  }

  if (active) {
    // D 16x16 f32 layout: VGPR r -> (M = r + 8*hi, N = nl)
#pragma unroll
    for (int t = 0; t < 4; ++t)
#pragma unroll
      for (int r = 0; r < 8; ++r)
        C[(size_t)(m0 + r + 8 * hi) * NC + (n0 + t * 16 + nl)] = acc[t][r];
  }
}

// ---------------- aggregation init: agg = bias + hw * dinv^2 (self-loop fused) ----------------
__global__ __launch_bounds__(256) void k_agg_init(const float* __restrict__ hw,
                                                  const float* __restrict__ dinv,
                                                  const float* __restrict__ bias,
                                                  float* __restrict__ agg, int n) {
  size_t idx = (size_t)blockIdx.x * 256 + threadIdx.x;  // one float4 per thread
  size_t total = (size_t)n * (HDIM / 4);
  if (idx >= total) return;
  int node = (int)(idx / (HDIM / 4));
  int j4 = (int)(idx % (HDIM / 4));
  float di = dinv[node];
  float w = di * di;
  float4 h = ((const float4*)hw)[idx];
  float4 b = ((const float4*)bias)[j4];
  float4 o;
  o.x = b.x + h.x * w;
  o.y = b.y + h.y * w;
  o.z = b.z + h.z * w;
  o.w = b.w + h.w * w;
  ((float4*)agg)[idx] = o;
}

// ---------------- edge scatter: agg[dst] += hw[src] * dinv[src]*dinv[dst] ----------------
// one wave32 per edge; 2 x float4 gather per lane, 8 fp32 atomics per lane.
__global__ __launch_bounds__(256) void k_scatter(const long long* __restrict__ src,
                                                 const long long* __restrict__ dst,
                                                 const float* __restrict__ dinv,
                                                 const float* __restrict__ hw,
                                                 float* __restrict__ agg, long long e) {
  long long ed = (long long)blockIdx.x * 8 + (threadIdx.x >> 5);
  if (ed >= e) return;
  int lane = threadIdx.x & 31;
  int s = (int)src[ed];
  int d = (int)dst[ed];
  float w = dinv[s] * dinv[d];
  const float4* hp = (const float4*)(hw + (size_t)s * HDIM) + lane * 2;
  float4 v0 = hp[0];
  float4 v1 = hp[1];
  float* ap = agg + (size_t)d * HDIM + lane * 8;
  unsafeAtomicAdd(ap + 0, v0.x * w);
  unsafeAtomicAdd(ap + 1, v0.y * w);
  unsafeAtomicAdd(ap + 2, v0.z * w);
  unsafeAtomicAdd(ap + 3, v0.w * w);
  unsafeAtomicAdd(ap + 4, v1.x * w);
  unsafeAtomicAdd(ap + 5, v1.y * w);
  unsafeAtomicAdd(ap + 6, v1.z * w);
  unsafeAtomicAdd(ap + 7, v1.w * w);
}

// ---------------- softmax over K=64, one wave32 per row ----------------
__global__ __launch_bounds__(256) void k_softmax(const float* __restrict__ logits,
                                                 float* __restrict__ S,
                                                 const float* __restrict__ log_tau, int n) {
  int row = blockIdx.x * 8 + (threadIdx.x >> 5);
  if (row >= n) return;
  int lane = threadIdx.x & 31;
  float inv_tau = __expf(-log_tau[0]);  // 1/tau = exp(-log_tau)
  const float* lp = logits + (size_t)row * KDIM;
  float x0 = lp[lane] * inv_tau;
  float x1 = lp[lane + 32] * inv_tau;
  float m = fmaxf(x0, x1);
#pragma unroll
  for (int o = 16; o > 0; o >>= 1) m = fmaxf(m, __shfl_xor(m, o, 32));
  float e0 = __expf(x0 - m);
  float e1 = __expf(x1 - m);
  float sum = e0 + e1;
#pragma unroll
  for (int o = 16; o > 0; o >>= 1) sum += __shfl_xor(sum, o, 32);
  float inv = 1.0f / sum;
  float* sp = S + (size_t)row * KDIM;
  sp[lane] = e0 * inv;
  sp[lane + 32] = e1 * inv;
}

extern "C" void kernel_launch(void* const* d_in, const int* in_sizes, int n_in,
                              void* d_out, int out_size, void* d_ws, size_t ws_size,
                              hipStream_t stream) {
  const float* x        = (const float*)d_in[0];
  const long long* ei   = (const long long*)d_in[1];  // int64 edge_index [2,E]
  const float* W1       = (const float*)d_in[2];
  const float* b1       = (const float*)d_in[3];
  const float* W2       = (const float*)d_in[4];
  const float* b2       = (const float*)d_in[5];
  const float* Wk       = (const float*)d_in[6];
  const float* log_tau  = (const float*)d_in[7];

  const int n       = in_sizes[0] / DDIM;
  const long long e = (long long)in_sizes[1] / 2;
  const long long* src = ei;
  const long long* dst = ei + e;

  float* S      = (float*)d_out;                 // tuple output: S first...
  float* logits = S + (size_t)n * KDIM;          // ...then logits

  // workspace carve
  char* p = (char*)d_ws;
  auto carve = [&](size_t bytes) -> void* {
    void* q = (void*)p;
    p += (bytes + 255) & ~(size_t)255;
    return q;
  };
  float* dinv = (float*)carve((size_t)n * sizeof(float));
  float* buf0 = (float*)carve((size_t)n * HDIM * sizeof(float));  // hw
  float* buf1 = (float*)carve((size_t)n * HDIM * sizeof(float));  // agg
  (void)ws_size; (void)n_in; (void)out_size;

  const int nb     = (n + 255) / 256;
  const int ebl256 = (int)((e + 255) / 256);
  const int ebl8   = (int)((e + 7) / 8);
  const size_t tot4 = (size_t)n * (HDIM / 4);
  const int nb4 = (int)((tot4 + 255) / 256);

  // degrees -> dinv (in place)
  k_init_deg<<<nb, 256, 0, stream>>>(dinv, n);
  k_deg_accum<<<ebl256, 256, 0, stream>>>(dst, dinv, e);
  k_dinv<<<nb, 256, 0, stream>>>(dinv, n);

  // layer 1: hw1 = x @ W1 ; agg1 = b1 + self + scatter
  dim3 g1((n + 127) / 128, HDIM / 64);
  k_gemm_wmma<DDIM, HDIM, false><<<g1, 256, 0, stream>>>(x, W1, buf0, n);
  k_agg_init<<<nb4, 256, 0, stream>>>(buf0, dinv, b1, buf1, n);
  k_scatter<<<ebl8, 256, 0, stream>>>(src, dst, dinv, buf0, buf1, e);

  // layer 2: hw2 = relu(agg1) @ W2 ; agg2 = b2 + self + scatter
  dim3 g2((n + 127) / 128, HDIM / 64);
  k_gemm_wmma<HDIM, HDIM, true><<<g2, 256, 0, stream>>>(buf1, W2, buf0, n);
  k_agg_init<<<nb4, 256, 0, stream>>>(buf0, dinv, b2, buf1, n);
  k_scatter<<<ebl8, 256, 0, stream>>>(src, dst, dinv, buf0, buf1, e);

  // logits = agg2 @ Wk  (written straight into d_out's second half)
  dim3 g3((n + 127) / 128, KDIM / 64);
  k_gemm_wmma<HDIM, KDIM, false><<<g3, 256, 0, stream>>>(buf1, Wk, logits, n);

  // S = softmax(logits / tau)
  k_softmax<<<(n + 7) / 8, 256, 0, stream>>>(logits, S, log_tau, n);
}